// Qwen2AttentionGrSlerpV5Wrapper_59511066853470
// MI455X (gfx1250) — compile-verified
//
#include <hip/hip_runtime.h>

#define BD 4
#define HD 8
#define SD 8192
#define DD 128
#define BHD (BD * HD)      // 32
#define STARTK 128
#define SUFB 7373
#define MID0 7245          // SUFB - STARTK
#define OUTSL 950          // 128 + 3 + 819
#define MIDF 3
#define EPSF 1e-8f
#define NITER 23
#define TILE_TOK 64

typedef unsigned int u32x4 __attribute__((ext_vector_type(4)));
typedef int i32x8 __attribute__((ext_vector_type(8)));
typedef int i32x4 __attribute__((ext_vector_type(4)));
typedef float v2f __attribute__((ext_vector_type(2)));
typedef float v8f __attribute__((ext_vector_type(8)));

// ---------------------------------------------------------------- helpers

__device__ __forceinline__ float clean1(float x) {
  // nan_to_num(nan=0, posinf=0, neginf=0) then clip(-100, 100)
  if (!(x == x)) x = 0.0f;
  if (fabsf(x) == __builtin_inff()) x = 0.0f;
  return fminf(fmaxf(x, -100.0f), 100.0f);
}

__device__ __forceinline__ void slerp2(float& ax, float& ay, float bx, float by, float t) {
  float na = fmaxf(sqrtf(ax * ax + ay * ay), EPSF);
  float nb = fmaxf(sqrtf(bx * bx + by * by), EPSF);
  float dot = (ax / na) * (bx / nb) + (ay / na) * (by / nb);
  dot = fminf(fmaxf(dot, -1.0f + 1e-6f), 1.0f - 1e-6f);
  float omega = acosf(dot);
  float sin_o = fmaxf(sinf(omega), EPSF);
  float fa = sinf((1.0f - t) * omega) / sin_o;
  float fb = sinf(t * omega) / sin_o;
  float lx = (1.0f - t) * ax + t * bx, ly = (1.0f - t) * ay + t * by;
  float sx = fa * ax + fb * bx,        sy = fa * ay + fb * by;
  float mx = (sin_o > 1e-4f) ? sx : lx;
  float my = (sin_o > 1e-4f) ? sy : ly;
  float sc = (1.0f - t) * na + t * nb;
  ax = clean1(mx * sc);
  ay = clean1(my * sc);
}

// -------------------------------------------------- TDM bulk copy (mid -> ws)
// One wave32 per 64-token tile: tensor_load_to_lds (global->LDS) then
// tensor_store_from_lds (LDS->global scratch). Pure DMA, no VALU on the data.
// D# packed per CDNA5 ISA 8.3/8.4: 2D tile, rows of 128 f32, data_size=4B.

__global__ __launch_bounds__(32) void k_copy_mid_tdm(const float* __restrict__ in,
                                                     float* __restrict__ C) {
  __shared__ float lds[TILE_TOK * DD];   // 32 KB staging
  const int tile = blockIdx.x;
  const int bh   = blockIdx.y;
  const int tok0 = tile * TILE_TOK;
  const int ntok = (MID0 - tok0) < TILE_TOK ? (MID0 - tok0) : TILE_TOK;

  unsigned long long src =
      (unsigned long long)(size_t)(in + ((size_t)bh * SD + STARTK + tok0) * DD);
  unsigned long long dst =
      (unsigned long long)(size_t)(C + ((size_t)bh * MID0 + tok0) * DD);
  // flat LDS addresses: low 32 bits are the LDS byte offset (aperture semantics)
  unsigned laddr = (unsigned)(size_t)(void*)lds;

  u32x4 g0;
  g0.x = 1u;                                                   // count=1, no gather
  g0.y = laddr;                                                // lds_addr
  g0.z = (unsigned)(src & 0xffffffffull);                      // global_addr[31:0]
  g0.w = (unsigned)((src >> 32) & 0x01ffffffull) | (2u << 30); // addr[56:32] | type=2

  i32x8 g1;
  g1[0] = (int)(2u << 16);                         // data_size = 2 (4 bytes)
  g1[1] = (int)(((unsigned)DD & 0xffffu) << 16);   // tensor_dim0 lo16 = 128
  g1[2] = (int)(((unsigned)ntok & 0xffffu) << 16); // dim0 hi16=0 | tensor_dim1 lo16
  g1[3] = (int)(((unsigned)DD & 0xffffu) << 16);   // dim1 hi16=0 | tile_dim0 = 128
  g1[4] = (int)((unsigned)ntok & 0xffffu);         // tile_dim1 = ntok, tile_dim2 = 0
  g1[5] = DD;                                      // tensor_dim0_stride lo32 = 128
  g1[6] = 0;
  g1[7] = 0;
  i32x4 z4 = {0, 0, 0, 0};
  i32x8 z8 = {0, 0, 0, 0, 0, 0, 0, 0};

  __builtin_amdgcn_tensor_load_to_lds(g0, g1, z4, z4, z8, 0);
  __builtin_amdgcn_s_wait_tensorcnt(0);

  u32x4 s0 = g0;
  s0.z = (unsigned)(dst & 0xffffffffull);
  s0.w = (unsigned)((dst >> 32) & 0x01ffffffull) | (2u << 30);
  __builtin_amdgcn_tensor_store_from_lds(s0, g1, z4, z4, z8, 0);
  __builtin_amdgcn_s_wait_tensorcnt(0);
}

// -------------------------------------------------------------- plan kernels

__global__ void k_init(const float* __restrict__ attn, float* __restrict__ sc,
                       int* __restrict__ idx) {
  int p = blockIdx.x * blockDim.x + threadIdx.x;
  if (p < MID0) { sc[p] = attn[STARTK + p]; idx[p] = p; }
}

// rank[p] = #{q : s[q] > s[p] || (s[q]==s[p] && q < p)}; kept iff rank < keepLen.
// Row-sums of the comparison matrix computed on the matrix pipe:
// D += A(16x4 of 0/1 cmp) x B(4x16 ones) via V_WMMA_F32_16X16X4_F32.
// Counts < 2^24 so f32 accumulation is exact. Each wave owns 16 p-rows;
// A layout: lanes 0-15 carry K=0,1 / lanes 16-31 carry K=2,3 (ISA 7.12.2);
// ranks read from lane 0 (M=0..7) and lane 16 (M=8..15) at N=0.
__global__ __launch_bounds__(256) void k_rank(const float* __restrict__ sc,
                                              int* __restrict__ kept,
                                              int L, int keepLen) {
  __shared__ float tile[256];
  int lane = threadIdx.x & 31;
  int wv   = threadIdx.x >> 5;                 // wave in block: 0..7
  int p0   = blockIdx.x * 128 + wv * 16;       // 16 p-rows per wave
  int m    = lane & 15;
  int p    = p0 + m;
  float sp = (p < L) ? sc[p] : 3.0e38f;        // sentinel: nothing outranks it
  int kb   = (lane < 16) ? 0 : 2;
  v8f acc = {0.f, 0.f, 0.f, 0.f, 0.f, 0.f, 0.f, 0.f};
  v2f ones; ones.x = 1.0f; ones.y = 1.0f;

  for (int base = 0; base < L; base += 256) {
    int q = base + threadIdx.x;
    tile[threadIdx.x] = (q < L) ? sc[q] : -3.0e38f;  // pad never counts
    __syncthreads();
    int lim = min(256, L - base);
    for (int qb = 0; qb < lim; qb += 4) {
      float s0 = tile[qb + kb];
      float s1 = tile[qb + kb + 1];
      int  q0 = base + qb + kb;
      v2f A;
      A.x = ((s0 > sp) || (s0 == sp && q0 < p)) ? 1.0f : 0.0f;
      A.y = ((s1 > sp) || (s1 == sp && (q0 + 1) < p)) ? 1.0f : 0.0f;
      acc = __builtin_amdgcn_wmma_f32_16x16x4_f32(false, A, false, ones,
                                                  (short)0, acc, false, false);
    }
    __syncthreads();
  }
  if (lane == 0 || lane == 16) {
    int mb = (lane == 0) ? 0 : 8;
    for (int v = 0; v < 8; ++v) {
      int pp = p0 + mb + v;
      if (pp < L) kept[pp] = (((int)acc[v]) < keepLen) ? 1 : 0;
    }
  }
}

// pos[p] = #kept q < p  -- same ones-matmul reduction on the matrix pipe
__global__ __launch_bounds__(256) void k_pos(const int* __restrict__ kept,
                                             int* __restrict__ pos, int L) {
  __shared__ float tile[256];
  int lane = threadIdx.x & 31;
  int wv   = threadIdx.x >> 5;
  int p0   = blockIdx.x * 128 + wv * 16;
  int m    = lane & 15;
  int p    = p0 + m;
  int kb   = (lane < 16) ? 0 : 2;
  v8f acc = {0.f, 0.f, 0.f, 0.f, 0.f, 0.f, 0.f, 0.f};
  v2f ones; ones.x = 1.0f; ones.y = 1.0f;

  for (int base = 0; base < L; base += 256) {
    int q = base + threadIdx.x;
    tile[threadIdx.x] = (q < L && kept[q]) ? 1.0f : 0.0f;
    __syncthreads();
    int lim = min(256, L - base);
    for (int qb = 0; qb < lim; qb += 4) {
      float s0 = tile[qb + kb];
      float s1 = tile[qb + kb + 1];
      int  q0 = base + qb + kb;
      v2f A;
      A.x = (s0 != 0.0f && q0 < p) ? 1.0f : 0.0f;
      A.y = (s1 != 0.0f && (q0 + 1) < p) ? 1.0f : 0.0f;
      acc = __builtin_amdgcn_wmma_f32_16x16x4_f32(false, A, false, ones,
                                                  (short)0, acc, false, false);
    }
    __syncthreads();
  }
  if (lane == 0 || lane == 16) {
    int mb = (lane == 0) ? 0 : 8;
    for (int v = 0; v < 8; ++v) {
      int pp = p0 + mb + v;
      if (pp < L) pos[pp] = (int)acc[v];
    }
  }
}

// compaction by index only: keepIdx (sorted), new scores base, new logical->slot map
__global__ void k_scatter(const int* __restrict__ kept, const int* __restrict__ pos,
                          const float* __restrict__ sc, const int* __restrict__ idx,
                          int* __restrict__ keepIdx, float* __restrict__ scNew,
                          int* __restrict__ idxNew, int L) {
  int p = blockIdx.x * blockDim.x + threadIdx.x;
  if (p >= L || !kept[p]) return;
  int k = pos[p];
  keepIdx[k] = p;
  scNew[k]   = sc[p];
  idxNew[k]  = idx[p];
}

// per-drop plan: anchor position/slot, drop slot, t, drop score
__global__ void k_drops(const int* __restrict__ kept, const int* __restrict__ pos,
                        const float* __restrict__ sc, const int* __restrict__ idx,
                        const int* __restrict__ keepIdx,
                        int* __restrict__ aPos, int* __restrict__ aSlot,
                        int* __restrict__ dSlot, float* __restrict__ tArr,
                        float* __restrict__ dSc, int L) {
  int p = blockIdx.x * blockDim.x + threadIdx.x;
  if (p >= L || kept[p]) return;
  int j    = p - pos[p];                  // ascending-drop-order index
  int apos = max(pos[p] - 1, 0);
  int aLog = keepIdx[apos];
  float a_s = sc[aLog], d_s = sc[p];
  float denom = a_s + d_s + EPSF;
  float t = (denom > EPSF) ? (a_s / denom) : 0.5f;
  aPos[j]  = apos;
  aSlot[j] = idx[aLog];
  dSlot[j] = idx[p];
  tArr[j]  = t;
  dSc[j]   = d_s;
}

// ------------------------------------------------------------- merge kernel
// block (a, bh): fold this anchor's contiguous drop run sequentially (matches
// the reference scan order); 64 lanes = one float-pair each, no cross-lane ops.

__global__ __launch_bounds__(64) void k_merge(float* __restrict__ C,
    const int* __restrict__ aPos, const int* __restrict__ aSlot,
    const int* __restrict__ dSlot, const float* __restrict__ tArr,
    const float* __restrict__ dSc, float* __restrict__ scNew, int nDrops) {
  int a  = blockIdx.x;
  int bh = blockIdx.y;
  int lo = 0, hb = nDrops;                 // lower_bound on sorted aPos
  while (lo < hb) { int m = (lo + hb) >> 1; if (aPos[m] < a) lo = m + 1; else hb = m; }
  if (lo >= nDrops || aPos[lo] != a) return;
  int hi = lo + 1;
  while (hi < nDrops && aPos[hi] == a) ++hi;

  int slot = aSlot[lo];
  float* abase = C + ((size_t)bh * MID0 + slot) * DD;
  int g = threadIdx.x;
  float ax = abase[2 * g], ay = abase[2 * g + 1];
  for (int j = lo; j < hi; ++j) {
    const float* dbase = C + ((size_t)bh * MID0 + dSlot[j]) * DD;
    if (j + 1 < hi)
      __builtin_prefetch(C + ((size_t)bh * MID0 + dSlot[j + 1]) * DD, 0, 0);
    slerp2(ax, ay, dbase[2 * g], dbase[2 * g + 1], tArr[j]);
  }
  abase[2 * g] = ax;
  abase[2 * g + 1] = ay;

  if (bh == 0 && g == 0) {                 // deterministic segment_sum (in drop order)
    float s = scNew[a];
    for (int j = lo; j < hi; ++j) s += dSc[j];
    scNew[a] = s;
  }
}

// ------------------------------------------------------------ output kernel

__global__ void k_out(const float* __restrict__ in, const float* __restrict__ C,
                      const int* __restrict__ idxF, float* __restrict__ out) {
  const int V = DD / 4;
  int tid = blockIdx.x * blockDim.x + threadIdx.x;
  int total = BHD * OUTSL * V;
  if (tid >= total) return;
  int v   = tid % V;
  int r   = tid / V;
  int tok = r % OUTSL;
  int bh  = r / OUTSL;
  const float* src;
  if (tok < STARTK) {
    src = in + ((size_t)bh * SD + tok) * DD + v * 4;
  } else if (tok < STARTK + MIDF) {
    src = C + ((size_t)bh * MID0 + idxF[tok - STARTK]) * DD + v * 4;
  } else {
    src = in + ((size_t)bh * SD + (tok + (SUFB - STARTK - MIDF))) * DD + v * 4;
  }
  float4 x = *(const float4*)src;
  x.x = clean1(x.x); x.y = clean1(x.y); x.z = clean1(x.z); x.w = clean1(x.w);
  *(float4*)(out + (size_t)tid * 4) = x;
}

// ------------------------------------------------------------------- launch

extern "C" void kernel_launch(void* const* d_in, const int* in_sizes, int n_in,
                              void* d_out, int out_size, void* d_ws, size_t ws_size,
                              hipStream_t stream) {
  (void)in_sizes; (void)n_in; (void)out_size; (void)ws_size;
  const float* states = (const float*)d_in[0];
  const float* attn   = (const float*)d_in[1];
  float* out = (float*)d_out;

  char* wsb = (char*)d_ws;
  size_t off = 0;
  auto carve = [&](size_t bytes) -> char* {
    char* p = wsb + off;
    off += (bytes + 255) & ~(size_t)255;
    return p;
  };
  float* C     = (float*)carve((size_t)BHD * MID0 * DD * sizeof(float)); // 118.7 MB
  float* scA   = (float*)carve(MID0 * sizeof(float));
  float* scB   = (float*)carve(MID0 * sizeof(float));
  int*   idxA  = (int*)carve(MID0 * sizeof(int));
  int*   idxB  = (int*)carve(MID0 * sizeof(int));
  int*   kept  = (int*)carve(MID0 * sizeof(int));
  int*   pos   = (int*)carve(MID0 * sizeof(int));
  int*   kIdx  = (int*)carve(MID0 * sizeof(int));
  int*   aPos  = (int*)carve(MID0 * sizeof(int));
  int*   aSlot = (int*)carve(MID0 * sizeof(int));
  int*   dSlot = (int*)carve(MID0 * sizeof(int));
  float* tArr  = (float*)carve(MID0 * sizeof(float));
  float* dSc   = (float*)carve(MID0 * sizeof(float));

  // fixed iteration schedule (ceil(L*0.7) under float64, matches reference)
  static const int LEN_TAB[NITER + 1] = {7245, 5072, 3551, 2486, 1741, 1219, 854,
                                         598, 419, 294, 206, 145, 102, 72, 51, 36,
                                         26, 19, 14, 10, 7, 5, 4, 3};

  dim3 gTdm((MID0 + TILE_TOK - 1) / TILE_TOK, BHD);
  k_copy_mid_tdm<<<gTdm, dim3(32), 0, stream>>>(states, C);
  k_init<<<(MID0 + 255) / 256, 256, 0, stream>>>(attn, scA, idxA);

  float* sCur = scA; float* sNxt = scB;
  int*   iCur = idxA; int*  iNxt = idxB;
  for (int it = 0; it < NITER; ++it) {
    int L = LEN_TAB[it], K = LEN_TAB[it + 1], ND = L - K;
    int gbW = (L + 127) / 128;   // WMMA kernels: 128 p-rows per 256-thread block
    int gb  = (L + 255) / 256;
    k_rank<<<gbW, 256, 0, stream>>>(sCur, kept, L, K);
    k_pos<<<gbW, 256, 0, stream>>>(kept, pos, L);
    k_scatter<<<gb, 256, 0, stream>>>(kept, pos, sCur, iCur, kIdx, sNxt, iNxt, L);
    k_drops<<<gb, 256, 0, stream>>>(kept, pos, sCur, iCur, kIdx,
                                    aPos, aSlot, dSlot, tArr, dSc, L);
    k_merge<<<dim3(K, BHD), 64, 0, stream>>>(C, aPos, aSlot, dSlot, tArr, dSc,
                                             sNxt, ND);
    float* ts = sCur; sCur = sNxt; sNxt = ts;
    int*   ti = iCur; iCur = iNxt; iNxt = ti;
  }

  int totalV = BHD * OUTSL * (DD / 4);
  k_out<<<(totalV + 255) / 256, 256, 0, stream>>>(states, C, iCur, out);
}